// ConditionalDisCoLoss_25022479467061
// MI455X (gfx1250) — compile-verified
//
#include <hip/hip_runtime.h>
#include <stdint.h>

#define NSAMP   8192
#define TILE    256
#define NBLK    32              // NSAMP / TILE
#define NCHUNK  8               // j-chunks for K3
#define TPC     (NBLK / NCHUNK) // tiles per chunk
#define THRESH  0.5f
#define EPS_F   1e-8f

typedef __attribute__((ext_vector_type(2))) float v2f;
typedef __attribute__((ext_vector_type(8))) float v8f;

// ---- workspace layout (float offsets) ----
#define WS_P      0
#define WS_RX     (NSAMP)
#define WS_RY     (2 * NSAMP)
#define PART_BCE  (3 * NSAMP)        // 32
#define PART_CNT  (PART_BCE + 32)    // 32
#define PART_SRX  (PART_BCE + 64)    // 32
#define PART_SRY  (PART_BCE + 96)    // 32
#define PART_SXY  (PART_BCE + 128)   // 256
#define PART_SXX  (PART_SXY + 256)   // 256
#define PART_SYY  (PART_SXX + 256)   // 256
#define SC_BASE   (PART_SYY + 256)
#define SC_INVC   (SC_BASE + 0)
#define SC_TOTX   (SC_BASE + 1)
#define SC_TOTY   (SC_BASE + 2)
#define SC_CNT    (SC_BASE + 3)

// ---------- CDNA5 async global -> LDS staging ----------
__device__ __forceinline__ void stage_async(float* dst_lds, const float* src_glb) {
    unsigned loff = (unsigned)(uintptr_t)dst_lds;              // low 32 bits = LDS offset
    unsigned long long g = (unsigned long long)(uintptr_t)src_glb;
    asm volatile("global_load_async_to_lds_b32 %0, %1, off"
                 :: "v"(loff), "v"(g) : "memory");
}
__device__ __forceinline__ void wait_async() {
    asm volatile("s_wait_asynccnt 0" ::: "memory");
}

// ---------- wave32 / block reductions ----------
__device__ __forceinline__ float wave_sum(float v) {
    #pragma unroll
    for (int off = 16; off > 0; off >>= 1)
        v += __shfl_down(v, off, 32);
    return v;
}

// valid at threadIdx.x == 0; smem must hold >= 8 floats
__device__ __forceinline__ float block_sum(float v, float* smem) {
    int lane = threadIdx.x & 31, wid = threadIdx.x >> 5;
    v = wave_sum(v);
    if (lane == 0) smem[wid] = v;
    __syncthreads();
    float r = 0.f;
    if (wid == 0) {
        r = (lane < (int)(blockDim.x >> 5)) ? smem[lane] : 0.f;
        r = wave_sum(r);
    }
    __syncthreads();
    return r;
}

// Block-reduce 256 values using the matrix unit.
// A = all-ones 16x4 => every row of D equals the column-sum vector of B.
// Each lane's two B VGPRs occupy two distinct slots of the 4x16 matrix, so
// sum(all B slots) == sum of all staged values per round. D VGPR0 across 32
// lanes covers rows {0,8} of D fully (documented layout), hence
// sum_lanes(d[0]) = 2 * total. Valid at tid == 0.
__device__ __forceinline__ float block_sum_wmma(float v, float* red, int tid) {
    __syncthreads();
    red[tid] = v;
    __syncthreads();
    float out = 0.f;
    if (tid < 32) {
        v2f a; a[0] = 1.0f; a[1] = 1.0f;
        v8f d = {};
        #pragma unroll
        for (int r = 0; r < 4; ++r) {
            v2f b;
            b[0] = red[tid + 64 * r];
            b[1] = red[tid + 64 * r + 32];
            d = __builtin_amdgcn_wmma_f32_16x16x4_f32(false, a, false, b,
                                                      (short)0, d, false, false);
        }
        float t = d[0];
        t = wave_sum(t);
        out = t * 0.5f;
    }
    __syncthreads();
    return out;
}

// ---------- K1: sigmoid, stable BCE, mask count ----------
__global__ void k1_sigmoid_bce(const float* __restrict__ x,
                               const float* __restrict__ t,
                               const float* __restrict__ s,
                               float* __restrict__ ws) {
    __shared__ float sm[8];
    int i = blockIdx.x * blockDim.x + threadIdx.x;
    float xi = x[i], ti = t[i], si = s[i];
    float bce = fmaxf(xi, 0.f) - xi * ti + log1pf(expf(-fabsf(xi)));
    ws[WS_P + i] = 1.f / (1.f + expf(-xi));
    float mi = (si >= THRESH) ? 1.f : 0.f;
    float bs = block_sum(bce, sm);
    float cs = block_sum(mi, sm);
    if (threadIdx.x == 0) {
        ws[PART_BCE + blockIdx.x] = bs;
        ws[PART_CNT + blockIdx.x] = cs;
    }
}

// ---------- K2: masked row sums rxs_i, rys_i ----------
__global__ void k2_rowsum(const float* __restrict__ s, float* __restrict__ ws) {
    __shared__ float p_t[TILE];
    __shared__ float s_t[TILE];
    __shared__ float sm[8];
    int tid = threadIdx.x;
    int i = blockIdx.x * TILE + tid;
    const float* p = ws + WS_P;
    float xi = p[i], yi = s[i];
    float rxs = 0.f, rys = 0.f;
    for (int tb = 0; tb < NBLK; ++tb) {
        __syncthreads();
        stage_async(&p_t[tid], p + tb * TILE + tid);
        stage_async(&s_t[tid], s + tb * TILE + tid);
        wait_async();
        __syncthreads();
        #pragma unroll 4
        for (int j = 0; j < TILE; ++j) {
            float pj = p_t[j], sj = s_t[j];
            float mj = (sj >= THRESH) ? 1.f : 0.f;
            rxs += mj * fabsf(xi - pj);
            rys += mj * fabsf(yi - sj);
        }
    }
    ws[WS_RX + i] = rxs;
    ws[WS_RY + i] = rys;
    float mi = (yi >= THRESH) ? 1.f : 0.f;
    float a = block_sum(mi * rxs, sm);
    float b = block_sum(mi * rys, sm);
    if (tid == 0) {
        ws[PART_SRX + blockIdx.x] = a;
        ws[PART_SRY + blockIdx.x] = b;
    }
}

// ---------- K2b: fold partials into scalars ----------
__global__ void k2b_scalars(float* __restrict__ ws) {
    int lane = threadIdx.x;
    float c   = wave_sum(ws[PART_CNT + lane]);
    float srx = wave_sum(ws[PART_SRX + lane]);
    float sry = wave_sum(ws[PART_SRY + lane]);
    if (lane == 0) {
        float csafe = fmaxf(c, 1.f);
        float invc  = 1.f / csafe;
        ws[SC_INVC] = invc;
        ws[SC_TOTX] = srx * invc * invc;
        ws[SC_TOTY] = sry * invc * invc;
        ws[SC_CNT]  = c;
    }
}

// ---------- K3: O(N^2) centered-product pass ----------
__global__ void k3_pairwise(const float* __restrict__ s, float* __restrict__ ws) {
    __shared__ float p_t[TILE], s_t[TILE], rx_t[TILE], ry_t[TILE];
    __shared__ float red[TILE];
    int tid = threadIdx.x;
    int i = blockIdx.x * TILE + tid;
    const float* p  = ws + WS_P;
    const float* rx = ws + WS_RX;
    const float* ry = ws + WS_RY;
    float invc = ws[SC_INVC], totx = ws[SC_TOTX], toty = ws[SC_TOTY];
    float xi = p[i], yi = s[i];
    float mi = (yi >= THRESH) ? 1.f : 0.f;
    float ca = totx - rx[i] * invc;  // cx = dx - rx_j*invc + ca
    float cb = toty - ry[i] * invc;
    float sxy = 0.f, sxx = 0.f, syy = 0.f;
    int tb0 = blockIdx.y * TPC;
    for (int tt = 0; tt < TPC; ++tt) {
        int tb = tb0 + tt;
        __syncthreads();
        stage_async(&p_t[tid],  p  + tb * TILE + tid);
        stage_async(&s_t[tid],  s  + tb * TILE + tid);
        stage_async(&rx_t[tid], rx + tb * TILE + tid);
        stage_async(&ry_t[tid], ry + tb * TILE + tid);
        wait_async();
        __syncthreads();
        #pragma unroll 4
        for (int j = 0; j < TILE; ++j) {
            float sj = s_t[j];
            float w  = mi * ((sj >= THRESH) ? 1.f : 0.f);
            float cx = fabsf(xi - p_t[j]) - rx_t[j] * invc + ca;
            float cy = fabsf(yi - sj)     - ry_t[j] * invc + cb;
            sxy += w * cx * cy;
            sxx += w * cx * cx;
            syy += w * cy * cy;
        }
    }
    float r0 = block_sum_wmma(sxy, red, tid);
    float r1 = block_sum_wmma(sxx, red, tid);
    float r2 = block_sum_wmma(syy, red, tid);
    if (tid == 0) {
        int part = blockIdx.y * gridDim.x + blockIdx.x;   // 0..255
        ws[PART_SXY + part] = r0;
        ws[PART_SXX + part] = r1;
        ws[PART_SYY + part] = r2;
    }
}

// ---------- K4: finalize ----------
__global__ void k4_final(float* __restrict__ ws, float* __restrict__ out) {
    __shared__ float sm[8];
    int tid = threadIdx.x;  // 256 threads
    float bce = (tid < 32) ? ws[PART_BCE + tid] : 0.f;
    float bs  = block_sum(bce, sm);
    float sxy = block_sum(ws[PART_SXY + tid], sm);
    float sxx = block_sum(ws[PART_SXX + tid], sm);
    float syy = block_sum(ws[PART_SYY + tid], sm);
    if (tid == 0) {
        float invc = ws[SC_INVC];
        float cnt  = ws[SC_CNT];
        float mm   = invc * invc;
        float dcov = sqrtf(fmaxf(sxy * mm, EPS_F));
        float dvx  = sqrtf(fmaxf(sxx * mm, EPS_F));
        float dvy  = sqrtf(fmaxf(syy * mm, EPS_F));
        float dcor = dcov / (dvx * dvy);
        out[0] = bs / (float)NSAMP + ((cnt > 0.f) ? dcor : 0.f);
    }
}

extern "C" void kernel_launch(void* const* d_in, const int* in_sizes, int n_in,
                              void* d_out, int out_size, void* d_ws, size_t ws_size,
                              hipStream_t stream) {
    (void)in_sizes; (void)n_in; (void)out_size; (void)ws_size;
    const float* x = (const float*)d_in[0];   // inputs  (8192,1) f32
    const float* t = (const float*)d_in[1];   // targets (8192,1) f32
    const float* s = (const float*)d_in[2];   // spectators (8192,) f32
    float* ws  = (float*)d_ws;
    float* out = (float*)d_out;

    k1_sigmoid_bce<<<NBLK, TILE, 0, stream>>>(x, t, s, ws);
    k2_rowsum     <<<NBLK, TILE, 0, stream>>>(s, ws);
    k2b_scalars   <<<1, 32, 0, stream>>>(ws);
    k3_pairwise   <<<dim3(NBLK, NCHUNK), TILE, 0, stream>>>(s, ws);
    k4_final      <<<1, TILE, 0, stream>>>(ws, out);
}